// HeteroGraphConvNet_35570919145822
// MI455X (gfx1250) — compile-verified
//
#include <hip/hip_runtime.h>

typedef __attribute__((ext_vector_type(2))) float v2f;
typedef __attribute__((ext_vector_type(8))) float v8f;

// ---------------- utility kernels ----------------

__global__ void k_zero(float* __restrict__ p, long n) {
  long i = (long)blockIdx.x * blockDim.x + threadIdx.x;
  if (i < n) p[i] = 0.0f;
}

// count degrees for all 3 relations: cbase holds [cs_b, cd_b, cs_r, cd_r, cs_j, cd_j], each N floats
__global__ void k_degrees(const int* __restrict__ sb, const int* __restrict__ db,
                          const int* __restrict__ sr, const int* __restrict__ dr,
                          const int* __restrict__ sj, const int* __restrict__ dj,
                          float* __restrict__ cbase, int nE, long N) {
  int e = blockIdx.x * blockDim.x + threadIdx.x;
  if (e >= nE) return;
  __hip_atomic_fetch_add(cbase + 0L * N + sb[e], 1.0f, __ATOMIC_RELAXED, __HIP_MEMORY_SCOPE_AGENT);
  __hip_atomic_fetch_add(cbase + 1L * N + db[e], 1.0f, __ATOMIC_RELAXED, __HIP_MEMORY_SCOPE_AGENT);
  __hip_atomic_fetch_add(cbase + 2L * N + sr[e], 1.0f, __ATOMIC_RELAXED, __HIP_MEMORY_SCOPE_AGENT);
  __hip_atomic_fetch_add(cbase + 3L * N + dr[e], 1.0f, __ATOMIC_RELAXED, __HIP_MEMORY_SCOPE_AGENT);
  __hip_atomic_fetch_add(cbase + 4L * N + sj[e], 1.0f, __ATOMIC_RELAXED, __HIP_MEMORY_SCOPE_AGENT);
  __hip_atomic_fetch_add(cbase + 5L * N + dj[e], 1.0f, __ATOMIC_RELAXED, __HIP_MEMORY_SCOPE_AGENT);
}

// c = 1/sqrt(max(c,1))
__global__ void k_norm(float* __restrict__ c, long n) {
  long i = (long)blockIdx.x * blockDim.x + threadIdx.x;
  if (i >= n) return;
  float d = fmaxf(c[i], 1.0f);
  c[i] = 1.0f / sqrtf(d);
}

// C[i] = b0[i&mask] + b1[i&mask] + b2[i&mask]  (bias of the 3 relations summed)
__global__ void k_init_bias3(float* __restrict__ C, const float* __restrict__ b0,
                             const float* __restrict__ b1, const float* __restrict__ b2,
                             long total, int colMask) {
  long i = (long)blockIdx.x * blockDim.x + threadIdx.x;
  if (i >= total) return;
  int j = (int)(i & (long)colMask);
  C[i] = b0[j] + b1[j] + b2[j];
}

// agg[dst[e]] += relu?(feat[src[e]]) * c_src[src[e]]   (float4 per thread, L2-resident f32 atomics)
__global__ void k_scatter(float* __restrict__ agg, const float* __restrict__ feat,
                          const int* __restrict__ src, const int* __restrict__ dst,
                          const float* __restrict__ csrc, long total, int logF4, int doRelu) {
  long tid = (long)blockIdx.x * blockDim.x + threadIdx.x;
  if (tid >= total) return;
  long e = tid >> logF4;
  int  j = (int)(tid & ((1L << logF4) - 1));
  int  s = src[e];
  int  d = dst[e];
  float c = csrc[s];
  int F = 4 << logF4;
  const float4* fp = (const float4*)(feat + (long)s * F);
  float4 v = fp[j];
  if (doRelu) {
    v.x = fmaxf(v.x, 0.0f); v.y = fmaxf(v.y, 0.0f);
    v.z = fmaxf(v.z, 0.0f); v.w = fmaxf(v.w, 0.0f);
  }
  float* o = agg + (long)d * F + 4 * j;
  __hip_atomic_fetch_add(o + 0, v.x * c, __ATOMIC_RELAXED, __HIP_MEMORY_SCOPE_AGENT);
  __hip_atomic_fetch_add(o + 1, v.y * c, __ATOMIC_RELAXED, __HIP_MEMORY_SCOPE_AGENT);
  __hip_atomic_fetch_add(o + 2, v.z * c, __ATOMIC_RELAXED, __HIP_MEMORY_SCOPE_AGENT);
  __hip_atomic_fetch_add(o + 3, v.w * c, __ATOMIC_RELAXED, __HIP_MEMORY_SCOPE_AGENT);
}

// ---------------- WMMA f32 GEMM: C[rows,cols] += diag(cdst) * A[rows,K] @ W[K,cols] ----------------
// One wave computes a 16(M) x 64(N) slab of C using V_WMMA_F32_16X16X4_F32 (full f32 precision).
// A-fragment layout (32-bit A 16x4): lane holds A[lane&15, 2*(lane>>4)+v], v = vgpr index.
// B-fragment layout mirrored:        lane holds W[2*(lane>>4)+v, N = lane&15].
// C/D layout: vgpr v, lanes 0-15 -> M=v, lanes 16-31 -> M=v+8; N = lane&15.
__global__ void k_wmma_gemm_acc(float* __restrict__ C, const float* __restrict__ A,
                                const float* __restrict__ W, const float* __restrict__ cdst,
                                int K, int cols, int colGroups, long totalWaves) {
  int  lane = threadIdx.x & 31;
  long wid  = (long)blockIdx.x * (blockDim.x >> 5) + (threadIdx.x >> 5);
  if (wid >= totalWaves) return;          // wave-uniform: EXEC stays all-ones for WMMA
  int rowTile = (int)(wid / colGroups);
  int colG    = (int)(wid % colGroups);
  int row0 = rowTile << 4;
  int col0 = colG << 6;
  int m    = lane & 15;
  int ksel = lane >> 4;

  const float* Ap = A + (long)(row0 + m) * K + 2 * ksel;
  const float* Wp = W + (long)(2 * ksel) * cols + col0 + m;

  v8f acc0 = {}, acc1 = {}, acc2 = {}, acc3 = {};

#pragma unroll 2
  for (int k = 0; k < K; k += 4) {
    v2f a;
    a.x = Ap[k];
    a.y = Ap[k + 1];
    const float* wk = Wp + (long)k * cols;
    v2f b0, b1, b2, b3;
    b0.x = wk[0];   b0.y = wk[cols + 0];
    b1.x = wk[16];  b1.y = wk[cols + 16];
    b2.x = wk[32];  b2.y = wk[cols + 32];
    b3.x = wk[48];  b3.y = wk[cols + 48];
    acc0 = __builtin_amdgcn_wmma_f32_16x16x4_f32(false, a, false, b0, (short)0, acc0, false, false);
    acc1 = __builtin_amdgcn_wmma_f32_16x16x4_f32(false, a, false, b1, (short)0, acc1, false, false);
    acc2 = __builtin_amdgcn_wmma_f32_16x16x4_f32(false, a, false, b2, (short)0, acc2, false, false);
    acc3 = __builtin_amdgcn_wmma_f32_16x16x4_f32(false, a, false, b3, (short)0, acc3, false, false);
  }

  // epilogue: scale rows by c_dst (linear in rows => equivalent to pre-scaling A), accumulate into C
#pragma unroll
  for (int v = 0; v < 8; ++v) {
    int   row = row0 + v + 8 * ksel;
    float cd  = cdst[row];
    float* cp = C + (long)row * cols + col0 + m;
    cp[0]  += cd * acc0[v];
    cp[16] += cd * acc1[v];
    cp[32] += cd * acc2[v];
    cp[48] += cd * acc3[v];
  }
}

// ---------------- host side ----------------

static inline unsigned cdiv(long a, long b) { return (unsigned)((a + b - 1) / b); }

extern "C" void kernel_launch(void* const* d_in, const int* in_sizes, int n_in,
                              void* d_out, int out_size, void* d_ws, size_t ws_size,
                              hipStream_t stream) {
  const float* x  = (const float*)d_in[0];
  const int* S[3] = { (const int*)d_in[1], (const int*)d_in[3], (const int*)d_in[5] };
  const int* D[3] = { (const int*)d_in[2], (const int*)d_in[4], (const int*)d_in[6] };
  // dict order: per relation r: W1_r, b1_r, W2_r, b2_r  (r = b, r, j)
  const float* W1[3] = { (const float*)d_in[7],  (const float*)d_in[11], (const float*)d_in[15] };
  const float* B1[3] = { (const float*)d_in[8],  (const float*)d_in[12], (const float*)d_in[16] };
  const float* W2[3] = { (const float*)d_in[9],  (const float*)d_in[13], (const float*)d_in[17] };
  const float* B2[3] = { (const float*)d_in[10], (const float*)d_in[14], (const float*)d_in[18] };

  const int  HID = in_sizes[8];                 // 256
  const int  OUT = in_sizes[10];                // 128
  const int  IN  = in_sizes[7] / HID;           // 128
  const long N   = (long)in_sizes[0] / IN;      // 100000
  const int  E   = in_sizes[1];                 // 500000

  float* out = (float*)d_out;

  // workspace layout
  float* cnorm = (float*)d_ws;                  // 6*N : cs_b, cd_b, cs_r, cd_r, cs_j, cd_j
  float* h     = cnorm + 6 * N;                 // N*HID
  float* agg   = h + N * (long)HID;             // N*HID (reused per relation; layer1 uses N*IN of it)

  float* cs[3] = { cnorm + 0 * N, cnorm + 2 * N, cnorm + 4 * N };
  float* cd[3] = { cnorm + 1 * N, cnorm + 3 * N, cnorm + 5 * N };

  const int B = 256;

  // --- degree normalization factors ---
  k_zero<<<cdiv(6 * N, B), B, 0, stream>>>(cnorm, 6 * N);
  k_degrees<<<cdiv(E, B), B, 0, stream>>>(S[0], D[0], S[1], D[1], S[2], D[2], cnorm, E, N);
  k_norm<<<cdiv(6 * N, B), B, 0, stream>>>(cnorm, 6 * N);

  // --- layer 1: h = sum_r diag(cd_r) * (scatter_r(x * cs_r)) @ W1_r + sum_r b1_r ---
  k_init_bias3<<<cdiv(N * HID, B), B, 0, stream>>>(h, B1[0], B1[1], B1[2], N * HID, HID - 1);
  {
    const int logF4 = 31 - __builtin_clz((unsigned)(IN / 4));   // IN=128 -> 5
    const long scatTot = (long)E * (IN / 4);
    const long waves   = (N / 16) * (long)(HID / 64);
    for (int r = 0; r < 3; ++r) {
      k_zero<<<cdiv(N * IN, B), B, 0, stream>>>(agg, N * (long)IN);
      k_scatter<<<cdiv(scatTot, B), B, 0, stream>>>(agg, x, S[r], D[r], cs[r], scatTot, logF4, 0);
      k_wmma_gemm_acc<<<cdiv(waves, 8), B, 0, stream>>>(h, agg, W1[r], cd[r], IN, HID, HID / 64, waves);
    }
  }

  // --- layer 2: out = sum_r diag(cd_r) * (scatter_r(relu(h) * cs_r)) @ W2_r + sum_r b2_r ---
  k_init_bias3<<<cdiv(N * OUT, B), B, 0, stream>>>(out, B2[0], B2[1], B2[2], N * OUT, OUT - 1);
  {
    const int logF4 = 31 - __builtin_clz((unsigned)(HID / 4));  // HID=256 -> 6
    const long scatTot = (long)E * (HID / 4);
    const long waves   = (N / 16) * (long)(OUT / 64);
    for (int r = 0; r < 3; ++r) {
      k_zero<<<cdiv(N * HID, B), B, 0, stream>>>(agg, N * (long)HID);
      k_scatter<<<cdiv(scatTot, B), B, 0, stream>>>(agg, h, S[r], D[r], cs[r], scatTot, logF4, 1);
      k_wmma_gemm_acc<<<cdiv(waves, 8), B, 0, stream>>>(out, agg, W2[r], cd[r], HID, OUT, OUT / 64, waves);
    }
  }
}